// ConformalSheafLearner_84834194030861
// MI455X (gfx1250) — compile-verified
//
#include <hip/hip_runtime.h>
#include <math.h>

// ---------------------------------------------------------------------------
// ConformalSheafLearner forward for MI455X (gfx1250, wave32).
//
//   h  = gelu(x @ emb1_w^T + emb1_b)
//   h  = gelu(gelu(h @ ws1^T + segsum(h[src],dst) @ wn1^T))
//   h  = gelu(gelu(h @ ws2^T + segsum(h[src],dst) @ wn2^T))
//   m  = h @ emb2_w^T + emb2_b ; out = (tanh(m[:, :2]), exp(min(m[:,2], ln10)))
//
// GEMMs use V_WMMA_F32_16X16X4_F32 (fp32-exact matrix path).
// Edge scatter uses per-lane global_atomic_add_f32 (L2-resident working set).
// ---------------------------------------------------------------------------

typedef __attribute__((ext_vector_type(2))) float v2f;
typedef __attribute__((ext_vector_type(8))) float v8f;

#define HID 128

__device__ __forceinline__ float gelu_exact(float x) {
  // torch default exact gelu: 0.5*x*(1+erf(x/sqrt(2)))
  return 0.5f * x * (1.0f + erff(x * 0.70710678118654752440f));
}

// ---------------------------------------------------------------------------
// WMMA fragment conventions (ISA 7.12.2, f32 16x16x4):
//   A (16x4, MxK): lane l(0..15) row M=r0+l holds K = kb+{0,1}; lanes 16..31
//                  hold K = kb+{2,3}.   -> one 8-byte load per lane.
//   B (4x16, KxN): lane l holds col N=n0+l, values W[N][kb+2*half + {0,1}]
//                  (B = W^T since Y = X @ W^T).
//   C/D (16x16):   acc[j] at lane l(+16*half) -> row r0+j+8*half, col n0+l.
// Each wave produces a 16x128 slab: 8 column tiles, 32 k-steps, 256 WMMAs.
// ---------------------------------------------------------------------------

// Y = gelu(X @ W^T + bias)
__global__ __launch_bounds__(256) void k_gemm_bias_gelu(
    const float* __restrict__ X, const float* __restrict__ W,
    const float* __restrict__ bias, float* __restrict__ Y, int n) {
  const int lane = threadIdx.x & 31;
  const int wv   = threadIdx.x >> 5;
  const int half = lane >> 4;
  const int l    = lane & 15;
  const int tile = blockIdx.x * 8 + wv;                // wave-uniform
  const int ntiles = (n + 15) >> 4;
  if (tile >= ntiles) return;                          // uniform exit: EXEC stays full
  const int r0 = tile * 16;
  int arow = r0 + l; if (arow >= n) arow = n - 1;      // clamp (keeps EXEC full)
  const float* ap = X + (size_t)arow * HID;

  v8f acc[8] = {};
  for (int kb = 0; kb < HID; kb += 4) {
    v2f a = *(const v2f*)(ap + kb + 2 * half);
#pragma unroll
    for (int nt = 0; nt < 8; ++nt) {
      const int col = nt * 16 + l;
      v2f b = *(const v2f*)(W + (size_t)col * HID + kb + 2 * half);
      acc[nt] = __builtin_amdgcn_wmma_f32_16x16x4_f32(
          false, a, false, b, (short)0, acc[nt], false, false);
    }
  }
#pragma unroll
  for (int nt = 0; nt < 8; ++nt) {
    const int col = nt * 16 + l;
    const float bv = bias[col];
#pragma unroll
    for (int j = 0; j < 8; ++j) {
      const int row = r0 + j + 8 * half;
      if (row < n) Y[(size_t)row * HID + col] = gelu_exact(acc[nt][j] + bv);
    }
  }
}

// Y = gelu(gelu(Hin @ Ws^T + Agg @ Wn^T))   (shared accumulators)
__global__ __launch_bounds__(256) void k_sumgnn_gemm(
    const float* __restrict__ Hin, const float* __restrict__ Agg,
    const float* __restrict__ Ws, const float* __restrict__ Wn,
    float* __restrict__ Y, int n) {
  const int lane = threadIdx.x & 31;
  const int wv   = threadIdx.x >> 5;
  const int half = lane >> 4;
  const int l    = lane & 15;
  const int tile = blockIdx.x * 8 + wv;
  const int ntiles = (n + 15) >> 4;
  if (tile >= ntiles) return;
  const int r0 = tile * 16;
  int arow = r0 + l; if (arow >= n) arow = n - 1;
  const float* hp = Hin + (size_t)arow * HID;
  const float* gp = Agg + (size_t)arow * HID;

  v8f acc[8] = {};
  for (int kb = 0; kb < HID; kb += 4) {
    v2f ah = *(const v2f*)(hp + kb + 2 * half);
    v2f ag = *(const v2f*)(gp + kb + 2 * half);
#pragma unroll
    for (int nt = 0; nt < 8; ++nt) {
      const int col = nt * 16 + l;
      v2f bs = *(const v2f*)(Ws + (size_t)col * HID + kb + 2 * half);
      v2f bn = *(const v2f*)(Wn + (size_t)col * HID + kb + 2 * half);
      acc[nt] = __builtin_amdgcn_wmma_f32_16x16x4_f32(
          false, ah, false, bs, (short)0, acc[nt], false, false);
      acc[nt] = __builtin_amdgcn_wmma_f32_16x16x4_f32(
          false, ag, false, bn, (short)0, acc[nt], false, false);
    }
  }
#pragma unroll
  for (int nt = 0; nt < 8; ++nt) {
    const int col = nt * 16 + l;
#pragma unroll
    for (int j = 0; j < 8; ++j) {
      const int row = r0 + j + 8 * half;
      if (row < n) Y[(size_t)row * HID + col] = gelu_exact(gelu_exact(acc[nt][j]));
    }
  }
}

// agg[dst[e]] += H[src[e]]  — one wave per edge: float4 gather + 4 f32 atomics/lane.
__global__ __launch_bounds__(256) void k_scatter_add(
    const float* __restrict__ Hsrc, const int* __restrict__ src,
    const int* __restrict__ dst, float* __restrict__ Agg, int nedges) {
  const int lane = threadIdx.x & 31;
  const int e = (int)((blockIdx.x * (unsigned)blockDim.x + threadIdx.x) >> 5);
  if (e >= nedges) return;
  const int s = src[e];                 // wave-uniform -> scalar broadcast
  const int d = dst[e];
  const float4 hv = *(const float4*)(Hsrc + (size_t)s * HID + lane * 4);
  float* base = Agg + (size_t)d * HID + lane * 4;
  __hip_atomic_fetch_add(base + 0, hv.x, __ATOMIC_RELAXED, __HIP_MEMORY_SCOPE_AGENT);
  __hip_atomic_fetch_add(base + 1, hv.y, __ATOMIC_RELAXED, __HIP_MEMORY_SCOPE_AGENT);
  __hip_atomic_fetch_add(base + 2, hv.z, __ATOMIC_RELAXED, __HIP_MEMORY_SCOPE_AGENT);
  __hip_atomic_fetch_add(base + 3, hv.w, __ATOMIC_RELAXED, __HIP_MEMORY_SCOPE_AGENT);
}

// m = h @ emb2_w^T + emb2_b ; out = (tanh(m0), tanh(m1), exp(min(m2, ln10)))
__global__ __launch_bounds__(256) void k_head(
    const float* __restrict__ Hin, const float* __restrict__ W2,
    const float* __restrict__ b2, float* __restrict__ out_tanh,
    float* __restrict__ out_exp, int n) {
  const int i = blockIdx.x * blockDim.x + threadIdx.x;
  if (i >= n) return;
  const float* hp = Hin + (size_t)i * HID;
  float m0 = b2[0], m1 = b2[1], m2 = b2[2];
#pragma unroll 4
  for (int k = 0; k < HID; k += 4) {
    const float4 hv = *(const float4*)(hp + k);
    m0 += hv.x * W2[k]           + hv.y * W2[k + 1]
        + hv.z * W2[k + 2]       + hv.w * W2[k + 3];
    m1 += hv.x * W2[HID + k]     + hv.y * W2[HID + k + 1]
        + hv.z * W2[HID + k + 2] + hv.w * W2[HID + k + 3];
    m2 += hv.x * W2[2*HID + k]     + hv.y * W2[2*HID + k + 1]
        + hv.z * W2[2*HID + k + 2] + hv.w * W2[2*HID + k + 3];
  }
  out_tanh[2 * i]     = tanhf(m0);
  out_tanh[2 * i + 1] = tanhf(m1);
  out_exp[i]          = expf(fminf(m2, 2.302585092994046f));  // ln(10)
}

extern "C" void kernel_launch(void* const* d_in, const int* in_sizes, int n_in,
                              void* d_out, int out_size, void* d_ws, size_t ws_size,
                              hipStream_t stream) {
  const float* x      = (const float*)d_in[0];
  const int*   ei     = (const int*)d_in[1];     // [2, E] (int32 per harness)
  const float* emb1_w = (const float*)d_in[2];
  const float* emb1_b = (const float*)d_in[3];
  const float* ws1    = (const float*)d_in[4];
  const float* wn1    = (const float*)d_in[5];
  const float* ws2    = (const float*)d_in[6];
  const float* wn2    = (const float*)d_in[7];
  const float* emb2_w = (const float*)d_in[8];
  const float* emb2_b = (const float*)d_in[9];

  const int n = in_sizes[0] / HID;               // 100000
  const int E = in_sizes[1] / 2;                 // 1600000
  const int* src = ei;
  const int* dst = ei + E;

  float* h0  = (float*)d_ws;                     // [n,128]
  float* agg = h0  + (size_t)n * HID;            // [n,128]
  float* h1  = agg + (size_t)n * HID;            // [n,128]

  float* out_tanh = (float*)d_out;               // [n,2]
  float* out_exp  = out_tanh + (size_t)n * 2;    // [n]

  const dim3 blk(256);
  const int row_tiles = (n + 15) / 16;
  const dim3 gemm_grid((row_tiles + 7) / 8);
  const dim3 edge_grid((E + 7) / 8);             // 8 waves per block, 1 edge/wave
  const dim3 head_grid((n + 255) / 256);

  // emb1 + gelu
  k_gemm_bias_gelu<<<gemm_grid, blk, 0, stream>>>(x, emb1_w, emb1_b, h0, n);

  // layer 0
  hipMemsetAsync(agg, 0, (size_t)n * HID * sizeof(float), stream);
  k_scatter_add<<<edge_grid, blk, 0, stream>>>(h0, src, dst, agg, E);
  k_sumgnn_gemm<<<gemm_grid, blk, 0, stream>>>(h0, agg, ws1, wn1, h1, n);

  // layer 1
  hipMemsetAsync(agg, 0, (size_t)n * HID * sizeof(float), stream);
  k_scatter_add<<<edge_grid, blk, 0, stream>>>(h1, src, dst, agg, E);
  k_sumgnn_gemm<<<gemm_grid, blk, 0, stream>>>(h1, agg, ws2, wn2, h0, n);

  // head + conformal output
  k_head<<<head_grid, blk, 0, stream>>>(h0, emb2_w, emb2_b, out_tanh, out_exp, n);
}